// AKConv_66632122630865
// MI455X (gfx1250) — compile-verified
//
#include <hip/hip_runtime.h>
#include <hip/hip_bf16.h>
#include <math.h>

typedef __attribute__((ext_vector_type(2))) float v2f;
typedef __attribute__((ext_vector_type(8))) float v8f;

#define Hc 128
#define Wc 128
#define HWc 16384
#define Cc 64
#define Oc 64
#define Pc 9
#define Kc 576          // C*P
#define BATCH 8

// ---------------------------------------------------------------------------
// Kernel 1: fused depthwise3x3 + BN + SiLU + pointwise(1x1) -> sample coords
// One thread per output pixel (b,y,x). Writes (sy,sx) for each of 9 points.
// ---------------------------------------------------------------------------
__global__ __launch_bounds__(256) void ak_offsets(
    const float* __restrict__ x,
    const float* __restrict__ dw_w, const float* __restrict__ dw_b,
    const float* __restrict__ bn_gamma, const float* __restrict__ bn_beta,
    const float* __restrict__ bn_mean, const float* __restrict__ bn_var,
    const float* __restrict__ pw_w, const float* __restrict__ pw_b,
    float* __restrict__ coords)
{
    int tid = blockIdx.x * blockDim.x + threadIdx.x;   // 0 .. B*HW-1
    int b   = tid >> 14;
    int pix = tid & (HWc - 1);
    int y   = pix >> 7;
    int xw  = pix & (Wc - 1);

    float off[2 * Pc];
#pragma unroll
    for (int j = 0; j < 2 * Pc; ++j) off[j] = pw_b[j];

    const float* xb = x + (size_t)b * Cc * HWc;

    for (int c = 0; c < Cc; ++c) {
        const float* xc = xb + (size_t)c * HWc;
        const float* wc = dw_w + c * 9;
        float acc = dw_b[c];
#pragma unroll
        for (int dy = -1; dy <= 1; ++dy) {
#pragma unroll
            for (int dx = -1; dx <= 1; ++dx) {
                int yy = y + dy, xx = xw + dx;
                float v = 0.0f;
                if (yy >= 0 && yy < Hc && xx >= 0 && xx < Wc)
                    v = xc[yy * Wc + xx];
                acc += v * wc[(dy + 1) * 3 + (dx + 1)];
            }
        }
        // BN (inference) + SiLU
        float inv = rsqrtf(bn_var[c] + 1e-5f);
        float h   = (acc - bn_mean[c]) * (bn_gamma[c] * inv) + bn_beta[c];
        float s   = h / (1.0f + __expf(-h));
        // accumulate pointwise conv (pw_w layout: [2P][C])
#pragma unroll
        for (int j = 0; j < 2 * Pc; ++j)
            off[j] += pw_w[j * Cc + c] * s;
    }

    // coords[(tid*9 + p)*2 + {0,1}] = (sy, sx); initial offsets are the 3x3
    // grid centered at 0: p = i*3+j -> (i-1, j-1). DILATION = 1.
    float* cp = coords + (size_t)tid * Pc * 2;
#pragma unroll
    for (int p = 0; p < Pc; ++p) {
        float dy0 = (float)(p / 3) - 1.0f;
        float dx0 = (float)(p % 3) - 1.0f;
        cp[2 * p + 0] = (float)y  + off[2 * p + 0] + dy0;
        cp[2 * p + 1] = (float)xw + off[2 * p + 1] + dx0;
    }
}

// ---------------------------------------------------------------------------
// Kernel 2: deformable gather + FP32 WMMA GEMM
//   out[b, n(pixel), o] = sum_{k=(c,p)} sampled[m][k] * W[k][o] + bias[o]
// Block = 128 threads = 4 waves. Each wave: 16 pixels x 64 outputs.
// Weight (576x64 fp32 = 147KB) staged in LDS once per block (320KB WGP LDS).
// ---------------------------------------------------------------------------
#define WBYTES   (Kc * Oc * 4)            // 147456
#define METAI_OFF WBYTES                  // int4  [4][144]
#define METAW_OFF (WBYTES + 4 * 144 * 16) // float4[4][144]
#define SMEM_TOTAL (WBYTES + 2 * 4 * 144 * 16)  // 165888 bytes

__global__ __launch_bounds__(128) void ak_gemm(
    const float* __restrict__ x,
    const float* __restrict__ weight,   // [O][C][P] = [O][K]
    const float* __restrict__ bias,
    const float* __restrict__ coords,
    float* __restrict__ out)
{
    extern __shared__ char smem[];
    float*  Wlds  = (float*)smem;                    // [K][O]
    int4*   metaI = (int4*)(smem + METAI_OFF);       // [wave][16*9]
    float4* metaW = (float4*)(smem + METAW_OFF);     // [wave][16*9]

    const int t    = threadIdx.x;
    const int wave = t >> 5;
    const int lane = t & 31;

    // ---- stage weight into LDS, transposed to [k][o] (coalesced global read)
    for (int i = t; i < Kc * Oc; i += 128) {
        int o = i / Kc;          // weight global is [o][k]
        int k = i - o * Kc;
        Wlds[k * Oc + o] = weight[i];
    }

    // ---- per-wave tile of 16 pixels
    const int tile      = blockIdx.x * 4 + wave;
    const int pixglobal = tile * 16;            // HW divisible by 16 -> one b
    const int b         = pixglobal >> 14;
    const int pix0      = pixglobal & (HWc - 1);

    // ---- precompute bilinear metadata (clamped indices, masked weights)
    int4*   mI = metaI + wave * 144;
    float4* mW = metaW + wave * 144;
    for (int e = lane; e < 16 * Pc; e += 32) {
        int m = e / Pc;
        int p = e - m * Pc;
        const float* cp = coords + ((size_t)(pixglobal + m) * Pc + p) * 2;
        float sy = cp[0], sx = cp[1];
        float fy0 = floorf(sy), fx0 = floorf(sx);
        int y0 = (int)fy0, x0 = (int)fx0;
        float wy1 = sy - fy0, wy0 = 1.0f - wy1;
        float wx1 = sx - fx0, wx0 = 1.0f - wx1;
        bool vy0 = (y0 >= 0) & (y0 < Hc);
        bool vy1 = (y0 + 1 >= 0) & (y0 + 1 < Hc);
        bool vx0 = (x0 >= 0) & (x0 < Wc);
        bool vx1 = (x0 + 1 >= 0) & (x0 + 1 < Wc);
        int yc0 = min(max(y0, 0), Hc - 1);
        int yc1 = min(max(y0 + 1, 0), Hc - 1);
        int xc0 = min(max(x0, 0), Wc - 1);
        int xc1 = min(max(x0 + 1, 0), Wc - 1);
        int4 ii;
        ii.x = yc0 * Wc + xc0;  ii.y = yc0 * Wc + xc1;
        ii.z = yc1 * Wc + xc0;  ii.w = yc1 * Wc + xc1;
        float4 ww;
        ww.x = (vy0 && vx0) ? wy0 * wx0 : 0.0f;
        ww.y = (vy0 && vx1) ? wy0 * wx1 : 0.0f;
        ww.z = (vy1 && vx0) ? wy1 * wx0 : 0.0f;
        ww.w = (vy1 && vx1) ? wy1 * wx1 : 0.0f;
        mI[e] = ii;
        mW[e] = ww;
    }
    __syncthreads();

    // ---- WMMA main loop over K = 576 in steps of 4
    // A 16x4 f32 layout (ISA 7.12.2): lane = m + 16*(k>>1), vgpr = k&1
    // C/D 16x16 f32: vgpr r -> M = r + 8*(lane>>4), N = lane&15
    const int m   = lane & 15;
    const int khi = lane >> 4;
    const int n   = lane & 15;
    const float* xb = x + (size_t)b * Cc * HWc;

    v8f acc[4] = {v8f{}, v8f{}, v8f{}, v8f{}};

    for (int kb = 0; kb < Kc; kb += 4) {
        int k0 = kb + 2 * khi;
        v2f a;
        {
            int c = k0 / Pc, p = k0 - c * Pc;
            int4   ii = mI[m * Pc + p];
            float4 ww = mW[m * Pc + p];
            const float* xc = xb + (size_t)c * HWc;
            a.x = ww.x * xc[ii.x] + ww.y * xc[ii.y] +
                  ww.z * xc[ii.z] + ww.w * xc[ii.w];
        }
        {
            int k1 = k0 + 1;
            int c = k1 / Pc, p = k1 - c * Pc;
            int4   ii = mI[m * Pc + p];
            float4 ww = mW[m * Pc + p];
            const float* xc = xb + (size_t)c * HWc;
            a.y = ww.x * xc[ii.x] + ww.y * xc[ii.y] +
                  ww.z * xc[ii.z] + ww.w * xc[ii.w];
        }
        // B 4x16 f32: vgpr 0 holds rows {k0}, vgpr 1 rows {k0+1} per half-wave
        const float* wr0 = Wlds + k0 * Oc;
        const float* wr1 = wr0 + Oc;
#pragma unroll
        for (int nt = 0; nt < 4; ++nt) {
            v2f bf;
            bf.x = wr0[nt * 16 + n];
            bf.y = wr1[nt * 16 + n];
            acc[nt] = __builtin_amdgcn_wmma_f32_16x16x4_f32(
                /*neg_a=*/false, a, /*neg_b=*/false, bf,
                /*c_mod=*/(short)0, acc[nt],
                /*reuse_a=*/false, /*reuse_b=*/false);
        }
    }

    // ---- epilogue: out[b][och][pix] = acc + bias  (out is B,O,H,W)
    float* outb = out + (size_t)b * Oc * HWc + pix0;
#pragma unroll
    for (int nt = 0; nt < 4; ++nt) {
        int och = nt * 16 + n;
        float bv = bias[och];
        float* op = outb + (size_t)och * HWc + 8 * khi;
#pragma unroll
        for (int r = 0; r < 8; ++r)
            op[r] = acc[nt][r] + bv;
    }
}

// ---------------------------------------------------------------------------
extern "C" void kernel_launch(void* const* d_in, const int* in_sizes, int n_in,
                              void* d_out, int out_size, void* d_ws, size_t ws_size,
                              hipStream_t stream) {
    const float* x        = (const float*)d_in[0];
    const float* dw_w     = (const float*)d_in[1];
    const float* dw_b     = (const float*)d_in[2];
    const float* bn_gamma = (const float*)d_in[3];
    const float* bn_beta  = (const float*)d_in[4];
    const float* bn_mean  = (const float*)d_in[5];
    const float* bn_var   = (const float*)d_in[6];
    const float* pw_w     = (const float*)d_in[7];
    const float* pw_b     = (const float*)d_in[8];
    const float* weight   = (const float*)d_in[9];
    const float* bias     = (const float*)d_in[10];
    float* out = (float*)d_out;

    float* coords = (float*)d_ws;   // B*HW*9*2 floats = 9.4 MB

    // allow >64KB dynamic LDS for the GEMM kernel (CDNA5: 320KB per WGP)
    hipFuncSetAttribute((const void*)ak_gemm,
                        hipFuncAttributeMaxDynamicSharedMemorySize, SMEM_TOTAL);

    // K1: one thread per pixel
    ak_offsets<<<(BATCH * HWc) / 256, 256, 0, stream>>>(
        x, dw_w, dw_b, bn_gamma, bn_beta, bn_mean, bn_var, pw_w, pw_b, coords);

    // K2: 4 waves/block, 16 pixels per wave -> B*HW/64 blocks
    ak_gemm<<<(BATCH * HWc) / 64, 128, SMEM_TOTAL, stream>>>(
        x, weight, bias, coords, out);
}